// Decoder_25288767439205
// MI455X (gfx1250) — compile-verified
//
#include <hip/hip_runtime.h>
#include <hip/hip_bf16.h>
#include <math.h>

#define Bdim 128
#define Ldim 2048
#define Hdim 256
#define Edim 256
#define GD   1024       // 4*H gate dim
#define SPLITS 8        // L-dimension split for attention phase 1
#define LSEG (Ldim / SPLITS)   // 256 rows per segment

typedef __attribute__((ext_vector_type(2))) float v2f;
typedef __attribute__((ext_vector_type(8))) float v8f;

// ---------------------------------------------------------------------------
// Kernel 1a: attention phase 1 — per-(batch, L-segment) online softmax partial.
// 1024 workgroups (B * SPLITS) to saturate HBM. 8 waves per block; each wave
// streams LSEG/8 rows; lane j owns H-chunk [8j, 8j+8). Emits unnormalized
// (m, s, ctx[256]) per block.
// ---------------------------------------------------------------------------
__global__ void attn_partial_kernel(const float* __restrict__ hidden1,
                                    const float* __restrict__ enc,
                                    float* __restrict__ part_m,
                                    float* __restrict__ part_s,
                                    float* __restrict__ part_ctx) {
    const int b    = blockIdx.x / SPLITS;
    const int sp   = blockIdx.x % SPLITS;
    const int tid  = threadIdx.x;
    const int wave = tid >> 5;
    const int lane = tid & 31;

    __shared__ float s_m[8];
    __shared__ float s_s[8];
    __shared__ float s_ctx[8 * Hdim];

    // hidden chunk for this lane (8 floats)
    const float* hb = hidden1 + (size_t)b * Hdim + lane * 8;
    float hv[8];
    {
        float4 h0 = *reinterpret_cast<const float4*>(hb);
        float4 h1 = *reinterpret_cast<const float4*>(hb + 4);
        hv[0]=h0.x; hv[1]=h0.y; hv[2]=h0.z; hv[3]=h0.w;
        hv[4]=h1.x; hv[5]=h1.y; hv[6]=h1.z; hv[7]=h1.w;
    }

    const float* encb = enc + (size_t)b * Ldim * Hdim;
    const int l0 = sp * LSEG;

    float m = -INFINITY;
    float s = 0.0f;
    float cacc[8];
#pragma unroll
    for (int j = 0; j < 8; ++j) cacc[j] = 0.0f;

    for (int l = l0 + wave; l < l0 + LSEG; l += 8) {
        const float* row = encb + (size_t)l * Hdim + lane * 8;
        __builtin_prefetch(row + 8 * Hdim, 0, 1);   // next row this wave reads
        float4 e0 = *reinterpret_cast<const float4*>(row);
        float4 e1 = *reinterpret_cast<const float4*>(row + 4);
        float er[8] = {e0.x, e0.y, e0.z, e0.w, e1.x, e1.y, e1.z, e1.w};

        float d = 0.0f;
#pragma unroll
        for (int j = 0; j < 8; ++j) d = fmaf(hv[j], er[j], d);
        // butterfly reduction across the wave32
        d += __shfl_xor(d, 16);
        d += __shfl_xor(d, 8);
        d += __shfl_xor(d, 4);
        d += __shfl_xor(d, 2);
        d += __shfl_xor(d, 1);

        float nm    = fmaxf(m, d);
        float scale = __expf(m - nm);   // first iter: exp(-inf) == 0
        float p     = __expf(d - nm);
        s = s * scale + p;
#pragma unroll
        for (int j = 0; j < 8; ++j) cacc[j] = fmaf(cacc[j], scale, p * er[j]);
        m = nm;
    }

    if (lane == 0) { s_m[wave] = m; s_s[wave] = s; }
#pragma unroll
    for (int j = 0; j < 8; ++j) s_ctx[wave * Hdim + lane * 8 + j] = cacc[j];
    __syncthreads();

    // combine the 8 per-wave partials; thread tid owns h = tid
    float M = -INFINITY;
#pragma unroll
    for (int w = 0; w < 8; ++w) M = fmaxf(M, s_m[w]);
    float S = 0.0f, acc = 0.0f;
#pragma unroll
    for (int w = 0; w < 8; ++w) {
        float sc = __expf(s_m[w] - M);
        S   = fmaf(s_s[w], sc, S);
        acc = fmaf(s_ctx[w * Hdim + tid], sc, acc);
    }
    const int pidx = b * SPLITS + sp;
    if (tid == 0) { part_m[pidx] = M; part_s[pidx] = S; }
    part_ctx[(size_t)pidx * Hdim + tid] = acc;   // unnormalized
}

// ---------------------------------------------------------------------------
// Kernel 1b: attention phase 2 — merge SPLITS partials per batch, normalize.
// ---------------------------------------------------------------------------
__global__ void attn_combine_kernel(const float* __restrict__ part_m,
                                    const float* __restrict__ part_s,
                                    const float* __restrict__ part_ctx,
                                    float* __restrict__ context) {
    const int b = blockIdx.x;
    const int h = threadIdx.x;

    float M = -INFINITY;
#pragma unroll
    for (int sp = 0; sp < SPLITS; ++sp)
        M = fmaxf(M, part_m[b * SPLITS + sp]);
    float S = 0.0f, acc = 0.0f;
#pragma unroll
    for (int sp = 0; sp < SPLITS; ++sp) {
        const int pidx = b * SPLITS + sp;
        float sc = __expf(part_m[pidx] - M);
        S   = fmaf(part_s[pidx], sc, S);
        acc = fmaf(part_ctx[(size_t)pidx * Hdim + h], sc, acc);
    }
    context[(size_t)b * Hdim + h] = acc / S;
}

// ---------------------------------------------------------------------------
// Kernel 2: generic LSTM gate GEMM via V_WMMA_F32_16X16X4_F32.
// gates(128x1024) = X0*W0^T + X1*W1^T + X2*W2^T + b_ih + b_hh
// One wave computes one 16x16 tile; K stepped by 4 per WMMA.
// A-matrix 16x4 layout (ISA 7.12.2): lanes 0-15 hold K=0,1; lanes 16-31 K=2,3.
// C/D: VGPR r holds M=r (lanes 0-15) and M=8+r (lanes 16-31), N=lane%16.
// ---------------------------------------------------------------------------
__device__ __forceinline__ void wmma_gemm_term(v8f& acc,
                                               const float* __restrict__ X, int ldx,
                                               const float* __restrict__ W, int ldw,
                                               int K, int m0, int n0, int lane) {
    if (X == nullptr) return;          // uniform scalar branch: EXEC stays all-1
    const int half = lane >> 4;        // 0: K+{0,1}   1: K+{2,3}
    const int l15  = lane & 15;
    const float* xr = X + (size_t)(m0 + l15) * ldx + 2 * half;
    const float* wr = W + (size_t)(n0 + l15) * ldw + 2 * half;
    for (int k0 = 0; k0 < K; k0 += 4) {
        v2f a, bm;
        a.x  = xr[k0];
        a.y  = xr[k0 + 1];
        bm.x = wr[k0];
        bm.y = wr[k0 + 1];
        acc = __builtin_amdgcn_wmma_f32_16x16x4_f32(
            /*neg_a=*/false, a, /*neg_b=*/false, bm,
            /*c_mod=*/(short)0, acc, /*reuse_a=*/false, /*reuse_b=*/false);
    }
}

__global__ void lstm_gates_wmma_kernel(float* __restrict__ gates,
                                       const float* __restrict__ b_ih,
                                       const float* __restrict__ b_hh,
                                       const float* __restrict__ X0, int ldx0,
                                       const float* __restrict__ W0, int ldw0, int K0,
                                       const float* __restrict__ X1, int ldx1,
                                       const float* __restrict__ W1, int ldw1, int K1,
                                       const float* __restrict__ X2, int ldx2,
                                       const float* __restrict__ W2, int ldw2, int K2) {
    const int lane   = threadIdx.x & 31;
    const int waveId = blockIdx.x * (blockDim.x >> 5) + (threadIdx.x >> 5);
    const int mt = waveId & 7;     // 8 tiles of M=128
    const int nt = waveId >> 3;    // 64 tiles of N=1024
    const int m0 = mt * 16;
    const int n0 = nt * 16;

    v8f acc = {};
    wmma_gemm_term(acc, X0, ldx0, W0, ldw0, K0, m0, n0, lane);
    wmma_gemm_term(acc, X1, ldx1, W1, ldw1, K1, m0, n0, lane);
    wmma_gemm_term(acc, X2, ldx2, W2, ldw2, K2, m0, n0, lane);

    const int half = lane >> 4;
    const int l15  = lane & 15;
    const int col  = n0 + l15;
    const float bias = b_ih[col] + b_hh[col];
#pragma unroll
    for (int r = 0; r < 8; ++r) {
        const int row = m0 + half * 8 + r;
        gates[(size_t)row * GD + col] = acc[r] + bias;
    }
}

// ---------------------------------------------------------------------------
// Kernel 3: LSTM pointwise gate recombination.
// ---------------------------------------------------------------------------
__device__ __forceinline__ float sigf(float x) {
    return 1.0f / (1.0f + __expf(-x));
}

__global__ void lstm_pointwise_kernel(const float* __restrict__ gates,
                                      const float* __restrict__ c_prev,
                                      float* __restrict__ h_out,
                                      float* __restrict__ h_out2,   // optional dup
                                      float* __restrict__ c_out) {
    const int idx = blockIdx.x * blockDim.x + threadIdx.x;
    if (idx >= Bdim * Hdim) return;
    const int b = idx >> 8;         // /H
    const int h = idx & (Hdim - 1); // %H
    const float* g = gates + (size_t)b * GD;
    const float i_g = g[h];
    const float f_g = g[Hdim + h];
    const float g_g = g[2 * Hdim + h];
    const float o_g = g[3 * Hdim + h];
    const float cp = c_prev[idx];
    const float cn = sigf(f_g) * cp + sigf(i_g) * tanhf(g_g);
    const float hn = sigf(o_g) * tanhf(cn);
    h_out[idx] = hn;
    if (h_out2) h_out2[idx] = hn;
    c_out[idx] = cn;
}

// ---------------------------------------------------------------------------
// Launcher
// ---------------------------------------------------------------------------
extern "C" void kernel_launch(void* const* d_in, const int* in_sizes, int n_in,
                              void* d_out, int out_size, void* d_ws, size_t ws_size,
                              hipStream_t stream) {
    const float* input_embed = (const float*)d_in[0];   // B x 1 x E
    const float* enc         = (const float*)d_in[1];   // B x L x H
    const float* hidden1     = (const float*)d_in[2];   // 1 x B x H
    const float* cell1       = (const float*)d_in[3];
    const float* hidden2     = (const float*)d_in[4];
    const float* cell2       = (const float*)d_in[5];
    const float* w_ih1       = (const float*)d_in[6];   // 1024 x 256
    const float* w_hh1       = (const float*)d_in[7];   // 1024 x 256
    const float* b_ih1       = (const float*)d_in[8];
    const float* b_hh1       = (const float*)d_in[9];
    const float* w_ih2       = (const float*)d_in[10];  // 1024 x 512
    const float* w_hh2       = (const float*)d_in[11];  // 1024 x 256
    const float* b_ih2       = (const float*)d_in[12];
    const float* b_hh2       = (const float*)d_in[13];

    float* out = (float*)d_out;
    const int BH = Bdim * Hdim;                 // 32768
    float* out_y  = out;                        // outputs (== h2)
    float* out_h1 = out + 1 * BH;
    float* out_c1 = out + 2 * BH;
    float* out_h2 = out + 3 * BH;
    float* out_c2 = out + 4 * BH;

    float* ws       = (float*)d_ws;
    float* ctx      = ws;                               // B x H
    float* gates1   = ws + BH;                          // B x 4H
    float* gates2   = ws + BH + Bdim * GD;              // B x 4H
    float* part_m   = ws + BH + 2 * Bdim * GD;          // B * SPLITS
    float* part_s   = part_m + Bdim * SPLITS;           // B * SPLITS
    float* part_ctx = part_s + Bdim * SPLITS;           // B * SPLITS * H

    // 1a) attention partials: 1024 blocks streaming 256 KB each (HBM-saturating)
    attn_partial_kernel<<<Bdim * SPLITS, 256, 0, stream>>>(
        hidden1, enc, part_m, part_s, part_ctx);

    // 1b) merge partials -> context
    attn_combine_kernel<<<Bdim, Hdim, 0, stream>>>(
        part_m, part_s, part_ctx, ctx);

    // 2) LSTM1 gates: ctx @ w_ih1^T + h1 @ w_hh1^T + b    (512 tiles / 4 waves per block)
    lstm_gates_wmma_kernel<<<128, 128, 0, stream>>>(
        gates1, b_ih1, b_hh1,
        ctx,     Hdim, w_ih1, Hdim, Hdim,
        hidden1, Hdim, w_hh1, Hdim, Hdim,
        nullptr, 0,    nullptr, 0,  0);

    // 3) LSTM1 pointwise -> h1, c1
    lstm_pointwise_kernel<<<(BH + 255) / 256, 256, 0, stream>>>(
        gates1, cell1, out_h1, nullptr, out_c1);

    // 4) LSTM2 gates: concat(embed,h1) @ w_ih2^T folded into two K=256 terms
    lstm_gates_wmma_kernel<<<128, 128, 0, stream>>>(
        gates2, b_ih2, b_hh2,
        input_embed, Edim, w_ih2,        Edim + Hdim, Edim,
        out_h1,      Hdim, w_ih2 + Edim, Edim + Hdim, Hdim,
        hidden2,     Hdim, w_hh2,        Hdim,        Hdim);

    // 5) LSTM2 pointwise -> h2 (dup into outputs), c2
    lstm_pointwise_kernel<<<(BH + 255) / 256, 256, 0, stream>>>(
        gates2, cell2, out_h2, out_y, out_c2);
}